// Create_Mask_23871428231714
// MI455X (gfx1250) — compile-verified
//
#include <hip/hip_runtime.h>
#include <cmath>

// mask[i][j] = ((i > j) | (i/n_nodes == j/n_nodes)) & (i != j)
// Since block_start(i) <= i < block_end(i), this reduces exactly to:
//   mask[i][j] = (j < be) && (j != i),   be = (i/n_nodes + 1) * n_nodes
//
// Row structure: a prefix of ones of length be with one zero at j == i.
// => 16-element chunks are all-zeros (j0 >= be), all-ones (fully inside the
//    prefix and not containing the diagonal), or mixed (<= 2 chunks per row).
//
// Store-bound kernel: 256 MB out @ 23.3 TB/s => ~11.5 us floor. Each lane
// writes 64 B via 4x global_store_b128 (non-temporal: output > 192 MB L2,
// never re-read). Bulk lanes do ~2 compares total; VALU is negligible.

typedef float v4f __attribute__((ext_vector_type(4)));

__global__ void __launch_bounds__(256)
create_mask_kernel(const int* __restrict__ n_nodes_p,
                   float* __restrict__ out,
                   int n) {
  const int i  = blockIdx.y;                                    // row (uniform)
  const int j0 = (blockIdx.x * blockDim.x + threadIdx.x) * 16;  // 16 cols/lane
  if (j0 >= n) return;

  const int n_nodes = *n_nodes_p;                 // uniform scalar load
  const int be = (i / n_nodes + 1) * n_nodes;     // block_end(i): SALU once/block

  float* __restrict__ row = out + (size_t)i * (size_t)n;

  if (j0 + 16 <= n) {
    v4f v0, v1, v2, v3;
    const bool all_zero = (j0 >= be);
    const bool all_ones = (j0 + 16 <= be) && ((unsigned)(i - j0) >= 16u);

    if (all_zero) {
      v0 = v1 = v2 = v3 = (v4f)0.0f;
    } else if (all_ones) {
      v0 = v1 = v2 = v3 = (v4f)1.0f;
    } else {
      // Mixed chunk: at most 2 per row (be boundary + diagonal).
      v4f vv[4];
#pragma unroll
      for (int q = 0; q < 4; ++q)
#pragma unroll
        for (int k = 0; k < 4; ++k) {
          const int j = j0 + q * 4 + k;
          vv[q][k] = ((j < be) && (j != i)) ? 1.0f : 0.0f;
        }
      v0 = vv[0]; v1 = vv[1]; v2 = vv[2]; v3 = vv[3];
    }

    v4f* p = (v4f*)(row + j0);
    __builtin_nontemporal_store(v0, p + 0);   // global_store_b128 th:NT
    __builtin_nontemporal_store(v1, p + 1);
    __builtin_nontemporal_store(v2, p + 2);
    __builtin_nontemporal_store(v3, p + 3);
  } else {
    // Tail (only if n % 16 != 0; not hit for n = 8192)
    for (int j = j0; j < n; ++j) {
      const float f = ((j < be) && (j != i)) ? 1.0f : 0.0f;
      __builtin_nontemporal_store(f, row + j);
    }
  }
}

extern "C" void kernel_launch(void* const* d_in, const int* in_sizes, int n_in,
                              void* d_out, int out_size, void* d_ws, size_t ws_size,
                              hipStream_t stream) {
  (void)in_sizes; (void)n_in; (void)d_ws; (void)ws_size;

  const int* n_nodes_p = (const int*)d_in[0];   // setup_inputs order: n_nodes, seq_len
  float* out = (float*)d_out;

  // out is n x n; recover n deterministically on the host (graph-capture safe).
  const int n = (int)(std::sqrt((double)out_size) + 0.5);

  const int lanes_per_row = (n + 15) / 16;        // 16 floats (64 B) per lane
  dim3 block(256, 1, 1);                          // 8 wave32 waves / block
  dim3 grid((unsigned)((lanes_per_row + 255) / 256), (unsigned)n, 1);

  create_mask_kernel<<<grid, block, 0, stream>>>(n_nodes_p, out, n);
}